// YvMamba3Block_72112500900652
// MI455X (gfx1250) — compile-verified
//
#include <hip/hip_runtime.h>
#include <hip/hip_bf16.h>
#include <math.h>

// Problem dims (fixed by reference setup_inputs)
#define Bb 2
#define Ss 2048
#define Mm 1024
#define Rr 64
#define Nn 16
#define CHUNK 64            // timesteps of B/C staged per LDS chunk

typedef __attribute__((ext_vector_type(2))) float v2f;
typedef __attribute__((ext_vector_type(8))) float v8f;
// Pointer element type demanded by __builtin_amdgcn_global_load_async_to_lds_b128
// (diagnosed from clang: "__attribute__((__vector_size__(4*sizeof(int)))) int *")
typedef __attribute__((__vector_size__(4 * sizeof(int)))) int v4i;

#if __has_builtin(__builtin_amdgcn_global_load_async_to_lds_b128)
#define HAVE_ASYNC_LDS 1
#else
#define HAVE_ASYNC_LDS 0
#endif

// Branchless softplus+clip: pure v_exp_f32/v_log_f32 + v_cndmask, EXEC untouched
// (important: lives in the same loop as WMMA, which requires EXEC all-1s).
__device__ __forceinline__ float softplus_clip(float x) {
    float e  = __expf(fminf(x, 20.0f));     // clamp arg so exp can't overflow
    float sp = __logf(1.0f + e);
    sp = (x > 20.0f) ? x : sp;              // select, not branch
    return fminf(fmaxf(sp, 1e-6f), 10.0f);
}

// ---------------------------------------------------------------------------
// Kernel 1: dt = clip(softplus(delta @ dt_proj_w^T + b), 1e-6, 10)
// One block = one 16-row s-tile; 4 waves split the 64 m-tiles.
// WMMA f32 16x16x4, K-loop over r=64 (16 steps). A-fragments preloaded once.
// ---------------------------------------------------------------------------
__global__ void __launch_bounds__(128)
dt_proj_kernel(const float* __restrict__ delta, const float* __restrict__ dtw,
               const float* __restrict__ dtb, float* __restrict__ dt) {
    const int tile = blockIdx.x;               // 0 .. Bb*Ss/16-1
    const int b    = tile / (Ss / 16);
    const int s0   = (tile % (Ss / 16)) * 16;
    const int lane = threadIdx.x & 31;
    const int wave = threadIdx.x >> 5;
    const int row  = lane & 15;                // M index
    const int koff = (lane >> 4) * 2;          // K sub-offset: 0 or 2

    // Preload entire 16x64 delta tile as 16 A-fragments (reused for all m-tiles)
    v2f afrag[16];
    const float* drow = delta + (size_t)(b * Ss + s0 + row) * Rr;
#pragma unroll
    for (int k = 0; k < 16; ++k) {
        afrag[k].x = drow[k * 4 + koff];
        afrag[k].y = drow[k * 4 + koff + 1];
    }

    const int col = lane & 15;                 // N index within m-tile
    for (int mt = wave; mt < Mm / 16; mt += 4) {
        const int m0 = mt * 16;
        const float* wrow = dtw + (size_t)(m0 + col) * Rr;  // dt_proj_w[m, r]
        v8f acc = {};
#pragma unroll
        for (int k = 0; k < 16; ++k) {
            v2f bfrag;
            bfrag.x = wrow[k * 4 + koff];
            bfrag.y = wrow[k * 4 + koff + 1];
            acc = __builtin_amdgcn_wmma_f32_16x16x4_f32(
                false, afrag[k], false, bfrag, (short)0, acc, false, false);
        }
        const float bias  = dtb[m0 + col];
        const int   rbase = s0 + ((lane >> 4) << 3);  // D: rows v (lanes<16) / v+8
#pragma unroll
        for (int vi = 0; vi < 8; ++vi) {
            float val = softplus_clip(acc[vi] + bias);
            dt[(size_t)(b * Ss + rbase + vi) * Mm + m0 + col] = val;
        }
    }
}

// ---------------------------------------------------------------------------
// Kernel 2: B/C projections: [16s x 1024m] x [1024m x 16n], WMMA f32 16x16x4.
// Two accumulators share each A-fragment (u) load.
// ---------------------------------------------------------------------------
__global__ void __launch_bounds__(32)
bc_proj_kernel(const float* __restrict__ u, const float* __restrict__ Bw,
               const float* __restrict__ Cw, float* __restrict__ Bout,
               float* __restrict__ Cout) {
    const int tile = blockIdx.x;
    const int b    = tile / (Ss / 16);
    const int s0   = (tile % (Ss / 16)) * 16;
    const int lane = threadIdx.x & 31;
    const int row  = lane & 15;
    const int koff = (lane >> 4) * 2;

    const float* urow  = u  + (size_t)(b * Ss + s0 + row) * Mm;
    const float* bwrow = Bw + (size_t)row * Mm;   // B_proj_w[n, m]; n == lane&15
    const float* cwrow = Cw + (size_t)row * Mm;

    v8f accB = {}, accC = {};
#pragma unroll 8
    for (int k = 0; k < Mm / 4; ++k) {
        v2f a, bb, cc;
        a.x  = urow[k * 4 + koff];   a.y  = urow[k * 4 + koff + 1];
        bb.x = bwrow[k * 4 + koff];  bb.y = bwrow[k * 4 + koff + 1];
        cc.x = cwrow[k * 4 + koff];  cc.y = cwrow[k * 4 + koff + 1];
        accB = __builtin_amdgcn_wmma_f32_16x16x4_f32(
            false, a, false, bb, (short)0, accB, false, false);
        accC = __builtin_amdgcn_wmma_f32_16x16x4_f32(
            false, a, false, cc, (short)0, accC, false, false);
    }
    const int rbase = s0 + ((lane >> 4) << 3);
    const int ncol  = lane & 15;
#pragma unroll
    for (int vi = 0; vi < 8; ++vi) {
        size_t o = (size_t)(b * Ss + rbase + vi) * Nn + ncol;
        Bout[o] = accB[vi];
        Cout[o] = accC[vi];
    }
}

// ---------------------------------------------------------------------------
// Kernel 3: fused selective scan + y contraction.
// Lane layout per wave32: lanes 0-15 -> n=0..15 for m0; lanes 16-31 -> m0+1.
// All 8 waves of a block share batch b, so B[b,t,:]/C[b,t,:] rows are staged
// once per block into LDS via async global->LDS copies (ASYNCcnt tracked),
// double-buffered in 64-timestep chunks.
// Recurrence (matches reference math exactly):
//   dA  = exp(dt*A); dA_c = max(dA, EPS)
//   S  += dt*B*u / dA_c ; P *= dA_c ; h = P*S
//   y   = sum_n h*C   (16-lane shfl_xor butterfly, never spills h to memory)
// ---------------------------------------------------------------------------
__global__ void __launch_bounds__(256)
scan_kernel(const float* __restrict__ u, const float* __restrict__ dt,
            const float* __restrict__ Bout, const float* __restrict__ Cout,
            const float* __restrict__ A_log, const float* __restrict__ D_skip,
            float* __restrict__ out, int* __restrict__ badflag) {
    const int gtid  = blockIdx.x * blockDim.x + threadIdx.x;
    const int w     = gtid >> 5;
    const int lane  = threadIdx.x & 31;
    const int n     = lane & 15;
    const int mpair = w % (Mm / 2);
    const int b     = w / (Mm / 2);
    const int m     = mpair * 2 + (lane >> 4);

    const float A   = fminf(fmaxf(-__expf(A_log[n * Mm + m]), -10.0f), -1e-6f);
    const float Dsk = D_skip[m];

    const float* ub  = u    + (size_t)b * Ss * Mm;
    const float* dtb = dt   + (size_t)b * Ss * Mm;
    const float* Bbp = Bout + (size_t)b * Ss * Nn;
    const float* Cbp = Cout + (size_t)b * Ss * Nn;
    float*       ob  = out  + (size_t)b * Ss * Mm;

    float P = 1.0f, Sacc = 0.0f;
    int bad = 0;

#if HAVE_ASYNC_LDS
    __shared__ float Bsh[2][CHUNK * Nn];
    __shared__ float Csh[2][CHUNK * Nn];
    const int tid = threadIdx.x;               // 256 threads x b128 = one chunk
    // Issue chunk 0 into buffer 0 (each thread copies 4 consecutive floats).
    __builtin_amdgcn_global_load_async_to_lds_b128(
        (v4i*)(Bbp + tid * 4), (v4i*)&Bsh[0][tid * 4], 0, 0);
    __builtin_amdgcn_global_load_async_to_lds_b128(
        (v4i*)(Cbp + tid * 4), (v4i*)&Csh[0][tid * 4], 0, 0);
    asm volatile("s_wait_asynccnt 0x0" ::: "memory");
    __syncthreads();
#endif

    for (int c = 0; c < Ss / CHUNK; ++c) {
#if HAVE_ASYNC_LDS
        if (c + 1 < Ss / CHUNK) {              // prefetch next chunk, other buffer
            const int nb = (c + 1) & 1;
            __builtin_amdgcn_global_load_async_to_lds_b128(
                (v4i*)(Bbp + (size_t)(c + 1) * CHUNK * Nn + tid * 4),
                (v4i*)&Bsh[nb][tid * 4], 0, 0);
            __builtin_amdgcn_global_load_async_to_lds_b128(
                (v4i*)(Cbp + (size_t)(c + 1) * CHUNK * Nn + tid * 4),
                (v4i*)&Csh[nb][tid * 4], 0, 0);
        }
        const float* __restrict__ Bl = Bsh[c & 1];
        const float* __restrict__ Cl = Csh[c & 1];
#endif
        for (int tt = 0; tt < CHUNK; ++tt) {
            const int t = c * CHUNK + tt;
            if (t + 8 < Ss) {  // global_prefetch_b8: hide sequential-t latency
                __builtin_prefetch(&ub[(size_t)(t + 8) * Mm + m], 0, 1);
                __builtin_prefetch(&dtb[(size_t)(t + 8) * Mm + m], 0, 1);
            }
            const float dtv = dtb[(size_t)t * Mm + m];
            const float uv  = ub[(size_t)t * Mm + m];
#if HAVE_ASYNC_LDS
            const float Bv  = Bl[tt * Nn + n];
            const float Cv  = Cl[tt * Nn + n];
#else
            const float Bv  = Bbp[t * Nn + n];
            const float Cv  = Cbp[t * Nn + n];
#endif
            const float dAc = fmaxf(__expf(dtv * A), 1e-10f);  // dA<=1 (A<0,dt>0)
            Sacc += (dtv * Bv * uv) / dAc;
            P    *= dAc;
            float contrib = P * Sacc * Cv;
#pragma unroll
            for (int o = 1; o < 16; o <<= 1)   // reduce over n within half-wave
                contrib += __shfl_xor(contrib, o, 32);
            if (n == 0) {
                float yv = contrib + uv * Dsk;
                if (!isfinite(yv)) bad = 1;
                yv = fminf(fmaxf(yv, -10000.0f), 10000.0f);
                ob[(size_t)t * Mm + m] = yv;
            }
        }
#if HAVE_ASYNC_LDS
        __syncthreads();                        // everyone done reading buf c&1
        if (c + 1 < Ss / CHUNK) {
            asm volatile("s_wait_asynccnt 0x0" ::: "memory");  // chunk c+1 landed
        }
        __syncthreads();                        // LDS writes visible block-wide
#endif
    }
    if (bad) atomicOr(badflag, 1);
}

__global__ void init_flag_kernel(int* __restrict__ f) { *f = 0; }

__global__ void fixup_kernel(float* __restrict__ out, const int* __restrict__ f,
                             int nElem) {
    int i = blockIdx.x * blockDim.x + threadIdx.x;
    if (i < nElem && *f) out[i] = 0.0f;
}

// ---------------------------------------------------------------------------
extern "C" void kernel_launch(void* const* d_in, const int* in_sizes, int n_in,
                              void* d_out, int out_size, void* d_ws, size_t ws_size,
                              hipStream_t stream) {
    const float* u      = (const float*)d_in[0];
    const float* delta  = (const float*)d_in[1];
    const float* dtw    = (const float*)d_in[2];
    const float* dtb    = (const float*)d_in[3];
    const float* A_log  = (const float*)d_in[4];
    const float* D_skip = (const float*)d_in[5];
    const float* Bw     = (const float*)d_in[6];
    const float* Cw     = (const float*)d_in[7];
    float* out = (float*)d_out;

    // Workspace layout: dt[b,s,m] | B[b,s,n] | C[b,s,n] | flag  (~17.3 MB)
    float* dt   = (float*)d_ws;
    float* Bout = dt + (size_t)Bb * Ss * Mm;
    float* Cout = Bout + (size_t)Bb * Ss * Nn;
    int*   flag = (int*)(Cout + (size_t)Bb * Ss * Nn);

    init_flag_kernel<<<1, 1, 0, stream>>>(flag);
    dt_proj_kernel<<<Bb * Ss / 16, 128, 0, stream>>>(delta, dtw, dtb, dt);
    bc_proj_kernel<<<Bb * Ss / 16, 32, 0, stream>>>(u, Bw, Cw, Bout, Cout);

    const int scanThreads = Bb * Nn * Mm;  // 32768 lanes = 1024 waves
    scan_kernel<<<scanThreads / 256, 256, 0, stream>>>(u, dt, Bout, Cout, A_log,
                                                       D_skip, out, flag);
    const int nElem = Bb * Ss * Mm;
    fixup_kernel<<<(nElem + 255) / 256, 256, 0, stream>>>(out, flag, nElem);
}